// BitStackLinear_70798240907324
// MI455X (gfx1250) — compile-verified
//
#include <hip/hip_runtime.h>
#include <cstdint>

// Problem constants (match reference)
#define TOKENS 2048
#define OUT_F  11008
#define IN_F   4096
#define RANK   16
#define WBIT   4

// Tiling
#define TM 256      // tokens per block
#define TN 128      // outputs per block
#define KI 32       // i-chunk per iteration
#define NT 256      // threads per block (8 wave32)
#define NCH (IN_F / KI)   // 128 chunks

// LDS strides (dwords): 16B-aligned rows, conflict-free frag reads
#define XS_STRIDE 36
#define WS_STRIDE 36
#define VT_STRIDE 36
#define US_STRIDE 20

// LDS layout (dword offsets); dynamic LDS base is offset 0 (no static __shared__)
#define XS0_OFF 0
#define XS_BUF  (TM * XS_STRIDE)                  // 9216 dwords per X buffer
#define WS_OFF  (XS0_OFF + 2 * XS_BUF)            // 18432
#define VT0_OFF (WS_OFF + TN * WS_STRIDE)         // 23040
#define VT_BUF  (WBIT * RANK * VT_STRIDE)         // 2304 dwords per VT buffer
#define US_OFF  (VT0_OFF + 2 * VT_BUF)            // 27648
#define QS0_OFF (US_OFF + WBIT * TN * US_STRIDE)  // 37888
#define QS_BUF  (WBIT * TN)                       // 512 dwords per Q buffer
#define SMEM_DW (QS0_OFF + 2 * QS_BUF)            // 38912 dwords
#define SMEM_BYTES (SMEM_DW * 4)                  // 155648 B (<320KB WGP LDS)

typedef float v2f __attribute__((ext_vector_type(2)));
typedef float v8f __attribute__((ext_vector_type(8)));

// D = A(16x4) * B(4x16) + C, all f32, wave32.
__device__ __forceinline__ v8f wmma4(v2f a, v2f b, v8f c) {
  return __builtin_amdgcn_wmma_f32_16x16x4_f32(false, a, false, b, (short)0, c, false, false);
}

// CDNA5 async global->LDS copy (GVS mode: saddr base + 32-bit vgpr offset),
// tracked with ASYNCcnt.
__device__ __forceinline__ void async_ld_b128(unsigned lds_byte, unsigned gbyte,
                                              unsigned long long base) {
  asm volatile("global_load_async_to_lds_b128 %0, %1, %2"
               :: "v"(lds_byte), "v"(gbyte), "s"(base) : "memory");
}
__device__ __forceinline__ void wait_async0() {
  asm volatile("s_wait_asynccnt 0x0" ::: "memory");
}

// Issue async staging of X tile (8 x b128/thread) and VT tile (2 x b128/thread)
// for chunk starting at column i0 into LDS buffer `buf`.
__device__ __forceinline__ void stage_async(const float* x, const float* vt,
                                            int tB, int i0, int buf, int tid) {
  const unsigned cx = (tid & 7) << 2;
  const unsigned rx = tid >> 3;
  const unsigned xbase = XS0_OFF + (unsigned)buf * XS_BUF;
  const unsigned long long xb = (unsigned long long)(uintptr_t)x;
  #pragma unroll
  for (int j = 0; j < 8; ++j) {
    unsigned row = rx + j * 32;
    unsigned lds = (xbase + row * XS_STRIDE + cx) * 4u;
    unsigned gof = (((unsigned)tB + row) * IN_F + (unsigned)i0 + cx) * 4u;
    async_ld_b128(lds, gof, xb);
  }
  const unsigned vbase = VT0_OFF + (unsigned)buf * VT_BUF;
  const unsigned long long vb = (unsigned long long)(uintptr_t)vt;
  #pragma unroll
  for (int j = 0; j < 2; ++j) {
    unsigned e = (unsigned)tid + (unsigned)j * NT;
    unsigned rowv = e >> 3, c = (e & 7) << 2;   // rowv == b*RANK+kk, vt is [64][IN_F]
    unsigned lds = (vbase + rowv * VT_STRIDE + c) * 4u;
    unsigned gof = (rowv * IN_F + (unsigned)i0 + c) * 4u;
    async_ld_b128(lds, gof, vb);
  }
}

struct QRegs { int4 a, b; };

__device__ __forceinline__ QRegs load_q(const int* qw, int oB, int ic, int tid) {
  QRegs r;
  const int e0 = tid, e1 = tid + NT;
  r.a = *(const int4*)(qw + (size_t)(e0 >> 7) * ((size_t)OUT_F * (IN_F / 8))
                          + (size_t)(oB + (e0 & 127)) * (IN_F / 8) + ic * 4);
  r.b = *(const int4*)(qw + (size_t)(e1 >> 7) * ((size_t)OUT_F * (IN_F / 8))
                          + (size_t)(oB + (e1 & 127)) * (IN_F / 8) + ic * 4);
  return r;
}

__device__ __forceinline__ unsigned pack_q(int4 q) {
  return (unsigned)(q.x & 0xFF) | ((unsigned)(q.y & 0xFF) << 8) |
         ((unsigned)(q.z & 0xFF) << 16) | ((unsigned)(q.w & 0xFF) << 24);
}

__device__ __forceinline__ void store_q(unsigned* Qdst, QRegs r, int tid) {
  const int e0 = tid, e1 = tid + NT;
  Qdst[(e0 >> 7) * TN + (e0 & 127)] = pack_q(r.a);
  Qdst[(e1 >> 7) * TN + (e1 & 127)] = pack_q(r.b);
}

__global__ __launch_bounds__(NT, 1)
void bitstack_fused_kernel(const float* __restrict__ x,
                           const int*   __restrict__ qw,
                           const float* __restrict__ u,
                           const float* __restrict__ vt,
                           float* __restrict__ y) {
  extern __shared__ float smem[];
  float*    Ws = smem + WS_OFF;
  float*    Us = smem + US_OFF;

  const int tid  = threadIdx.x;
  const int lane = tid & 31;
  const int w    = tid >> 5;        // wave 0..7
  const int hi   = lane >> 4;       // lane half
  const int l16  = lane & 15;

  const int oB = blockIdx.x * TN;
  const int tB = blockIdx.y * TM;

  // ---- preload U strip: u[b, oB..oB+TN, 0..15] -> Us (block-resident) ----
  #pragma unroll
  for (int j = 0; j < 8; ++j) {
    int e   = tid + j * NT;          // float4 unit index, 2048 total
    int b   = e >> 9;
    int rem = e & 511;
    int ol  = rem >> 2;
    int k4  = (rem & 3) << 2;
    float4 v = *(const float4*)(u + ((size_t)b * OUT_F + (size_t)(oB + ol)) * RANK + k4);
    *(float4*)(Us + (b * TN + ol) * US_STRIDE + k4) = v;
  }

  // ---- wave tiling: 4t x 2o wave grid, each wave a 64x64 C tile ----
  const int tw  = w >> 1, ow = w & 1;
  const int t0  = tw * 64;
  const int o0m = ow * 64;
  const int o0r = w * 16;           // recon o-tile for this wave

  v8f acc[4][4];
  #pragma unroll
  for (int tt = 0; tt < 4; ++tt)
    #pragma unroll
    for (int ot = 0; ot < 4; ++ot) { v8f z = {}; acc[tt][ot] = z; }

  // ---- prologue: stage chunk 0 ----
  stage_async(x, vt, tB, 0, 0, tid);
  {
    QRegs q0 = load_q(qw, oB, 0, tid);
    store_q((unsigned*)smem + QS0_OFF, q0, tid);
  }
  wait_async0();
  __syncthreads();

  for (int ic = 0; ic < NCH; ++ic) {
    const int cur = ic & 1, nxt = cur ^ 1;
    const bool have_next = (ic + 1) < NCH;

    // ---- prefetch chunk ic+1 while computing chunk ic ----
    QRegs qn;
    if (have_next) {
      stage_async(x, vt, tB, (ic + 1) * KI, nxt, tid);
      qn = load_q(qw, oB, ic + 1, tid);
    }

    const float*    VTc = smem + VT0_OFF + cur * VT_BUF;
    const float*    Xc  = smem + XS0_OFF + cur * XS_BUF;
    const unsigned* Qc  = (const unsigned*)smem + QS0_OFF + cur * QS_BUF;

    // ---- reconstruct W chunk [TN x KI] via WMMA; each wave: 2 subtiles ----
    #pragma unroll
    for (int s = 0; s < 2; ++s) {
      const int icol = s * 16 + l16;          // 0..31 within chunk
      v8f wacc = {};
      #pragma unroll
      for (int b = 0; b < WBIT; ++b) {
        // hoist all fragments + sign words before the WMMA chain
        const float* ub = Us  + (b * TN + o0r + l16) * US_STRIDE + 2 * hi;
        const float* vb = VTc + (b * RANK + 2 * hi) * VT_STRIDE + icol;
        v2f a[4], bb[4];
        unsigned qv[8];
        #pragma unroll
        for (int k = 0; k < 4; ++k) {
          a[k].x  = ub[4 * k];
          a[k].y  = ub[4 * k + 1];
          bb[k].x = vb[(4 * k    ) * VT_STRIDE];
          bb[k].y = vb[(4 * k + 1) * VT_STRIDE];
        }
        #pragma unroll
        for (int r = 0; r < 8; ++r) qv[r] = Qc[b * TN + o0r + 8 * hi + r];
        v8f d = {};
        #pragma unroll
        for (int k = 0; k < 4; ++k) d = wmma4(a[k], bb[k], d);
        // fold sign: bit==1 -> +1, bit==0 -> -1 (xor bit31)
        #pragma unroll
        for (int r = 0; r < 8; ++r) {
          unsigned neg = ((qv[r] >> icol) & 1u) ^ 1u;
          wacc[r] += __uint_as_float(__float_as_uint(d[r]) ^ (neg << 31));
        }
      }
      #pragma unroll
      for (int r = 0; r < 8; ++r)
        Ws[(o0r + 8 * hi + r) * WS_STRIDE + icol] = wacc[r];
    }
    __syncthreads();   // Ws ready

    // ---- main GEMM on chunk: C[64t x 64o] += X[64 x 32] * W^T[32 x 64] ----
    #pragma unroll
    for (int k4 = 0; k4 < KI; k4 += 4) {
      v2f Af[4], Bf[4];
      #pragma unroll
      for (int tt = 0; tt < 4; ++tt) {
        const float* p = Xc + (t0 + tt * 16 + l16) * XS_STRIDE + k4 + 2 * hi;
        Af[tt].x = p[0]; Af[tt].y = p[1];
      }
      #pragma unroll
      for (int ot = 0; ot < 4; ++ot) {
        const float* p = Ws + (o0m + ot * 16 + l16) * WS_STRIDE + k4 + 2 * hi;
        Bf[ot].x = p[0]; Bf[ot].y = p[1];
      }
      #pragma unroll
      for (int tt = 0; tt < 4; ++tt)
        #pragma unroll
        for (int ot = 0; ot < 4; ++ot)
          acc[tt][ot] = wmma4(Af[tt], Bf[ot], acc[tt][ot]);
    }

    // ---- commit next chunk's packed sign bits; drain async; flip buffers ----
    if (have_next)
      store_q((unsigned*)smem + QS0_OFF + nxt * QS_BUF, qn, tid);
    wait_async0();
    __syncthreads();
  }

  // ---- epilogue: C layout M = r + 8*hi, N = l16 ----
  #pragma unroll
  for (int tt = 0; tt < 4; ++tt)
    #pragma unroll
    for (int ot = 0; ot < 4; ++ot)
      #pragma unroll
      for (int r = 0; r < 8; ++r) {
        int tg = tB + t0 + tt * 16 + 8 * hi + r;
        int og = oB + o0m + ot * 16 + l16;
        y[(size_t)tg * OUT_F + og] = acc[tt][ot][r];
      }
}

extern "C" void kernel_launch(void* const* d_in, const int* in_sizes, int n_in,
                              void* d_out, int out_size, void* d_ws, size_t ws_size,
                              hipStream_t stream) {
  (void)in_sizes; (void)n_in; (void)out_size; (void)d_ws; (void)ws_size;
  const float* x  = (const float*)d_in[0];
  const int*   qw = (const int*)d_in[1];
  const float* u  = (const float*)d_in[2];
  const float* vt = (const float*)d_in[3];
  float* y = (float*)d_out;

  hipFuncSetAttribute(reinterpret_cast<const void*>(bitstack_fused_kernel),
                      hipFuncAttributeMaxDynamicSharedMemorySize, SMEM_BYTES);

  dim3 grid(OUT_F / TN, TOKENS / TM);   // 86 x 8
  bitstack_fused_kernel<<<grid, NT, SMEM_BYTES, stream>>>(x, qw, u, vt, y);
}